// QuantumVariationalLayer_51238959841322
// MI455X (gfx1250) — compile-verified
//
#include <hip/hip_runtime.h>
#include <hip/hip_bf16.h>
#include <math.h>

// ---------------------------------------------------------------------------
// QuantumVariationalLayer fused kernel for gfx1250 (MI455X, wave32, WMMA).
//   latent = dec2( gelu( dec1( quantum( enc(x) ) ) ) )  +  entropy accumulators
// FP32 WMMA (V_WMMA_F32_16X16X4_F32) keeps reference precision.
// ---------------------------------------------------------------------------

typedef float v2f __attribute__((ext_vector_type(2)));
typedef float v8f __attribute__((ext_vector_type(8)));

#define TOK_TOTAL (64 * 4096)      // B*S
#define LATENT_ELEMS ((size_t)TOK_TOTAL * 64)

static __device__ __forceinline__ v8f wmma4(v2f a, v2f b, v8f c) {
  // D = A(16x4) * B(4x16) + C(16x16), fp32
  return __builtin_amdgcn_wmma_f32_16x16x4_f32(
      /*neg_a=*/false, a, /*neg_b=*/false, b,
      /*c_mod=*/(short)0, c, /*reuse_a=*/false, /*reuse_b=*/false);
}

static __device__ __forceinline__ float gelu_exact(float x) {
  return 0.5f * x * (1.0f + erff(x * 0.70710678118654752f));
}

// Apply generic complex 2x2 gate on qubit with bitmask m (state in registers).
static __device__ __forceinline__ void apply1q(
    float* zr, float* zi, int m,
    float u00r, float u00i, float u01r, float u01i,
    float u10r, float u10i, float u11r, float u11i) {
#pragma unroll
  for (int s = 0; s < 16; ++s) {
    if (!(s & m)) {
      const int t = s | m;
      const float ar = zr[s], ai = zi[s];
      const float br = zr[t], bi = zi[t];
      zr[s] = u00r * ar - u00i * ai + u01r * br - u01i * bi;
      zi[s] = u00r * ai + u00i * ar + u01r * bi + u01i * br;
      zr[t] = u10r * ar - u10i * ai + u11r * br - u11i * bi;
      zi[t] = u10r * ai + u10i * ar + u11r * bi + u11i * br;
    }
  }
}

__global__ void qvl_zero_ent(float* ent) {
  if (threadIdx.x < 4) ent[threadIdx.x] = 0.0f;
}

__global__ __launch_bounds__(128) void qvl_fused(
    const float* __restrict__ x,        // (BS, 128)
    const float* __restrict__ enc_w1,   // (128, 64)
    const float* __restrict__ enc_b1,   // (64,)
    const float* __restrict__ enc_w2,   // (64, 8)
    const float* __restrict__ enc_b2,   // (8,)
    const float* __restrict__ q_params, // (3, 4, 3)
    const float* __restrict__ dec_w1,   // (4, 16)
    const float* __restrict__ dec_b1,   // (16,)
    const float* __restrict__ dec_w2,   // (16, 64)
    const float* __restrict__ dec_b2,   // (64,)
    float* __restrict__ out,            // (BS, 64)
    float* __restrict__ ent)            // (4,)
{
  // Weights staged transposed so B-fragments are contiguous float2 in K.
  __shared__ float s_w1t[64][128];   // enc_w1^T   32 KB
  __shared__ float s_w2t[8][64];     // enc_w2^T    2 KB
  __shared__ float s_dw1t[16][4];    // dec_w1^T
  __shared__ float s_dw2t[64][16];   // dec_w2^T    4 KB
  __shared__ float s_b1[64];
  __shared__ float s_b2[8];
  __shared__ float s_db1[16];
  __shared__ float s_db2[64];
  __shared__ float s_rot[12][8];     // per (layer,qubit) complex 2x2 Rot
  __shared__ float s_h[4][16][66];   // per-wave h tile (16x64, padded)
  __shared__ float s_ang[4][16][8];  // per-wave angles
  __shared__ float s_qf[4][16][4];   // per-wave PauliZ expvals
  __shared__ float s_g[4][16][18];   // per-wave gelu(dec1) tile (padded)
  __shared__ float s_ent[4];

  const int tid  = threadIdx.x;
  const int wv   = tid >> 5;        // wave id 0..3
  const int lane = tid & 31;
  const int col  = lane & 15;       // N-column within 16-wide tile / A-row M
  const int hi   = lane >> 4;       // half-wave select
  const int r0   = hi * 2;          // K sub-offset: lanes0-15 -> K0,K1; 16-31 -> K2,K3

  // ---- stage weights / biases / rot matrices --------------------------------
  for (int i = tid; i < 128 * 64; i += 128) s_w1t[i & 63][i >> 6] = enc_w1[i];
  for (int i = tid; i < 64 * 8;  i += 128) s_w2t[i & 7][i >> 3]  = enc_w2[i];
  for (int i = tid; i < 4 * 16;  i += 128) s_dw1t[i & 15][i >> 4] = dec_w1[i];
  for (int i = tid; i < 16 * 64; i += 128) s_dw2t[i & 63][i >> 6] = dec_w2[i];
  if (tid < 64) s_b1[tid] = enc_b1[tid];
  if (tid < 8)  s_b2[tid] = enc_b2[tid];
  if (tid < 16) s_db1[tid] = dec_b1[tid];
  if (tid >= 64 && tid < 128) s_db2[tid - 64] = dec_b2[tid - 64];
  if (tid < 4) s_ent[tid] = 0.0f;
  if (tid < 12) {
    // Rot(phi,theta,omega) = [[ep*c, -em*s],[conj(em)*s, conj(ep)*c]]
    const float phi = q_params[tid * 3 + 0];
    const float th  = q_params[tid * 3 + 1];
    const float om  = q_params[tid * 3 + 2];
    const float ch = cosf(0.5f * th), sh = sinf(0.5f * th);
    const float pa = -0.5f * (phi + om);     // ep = (cos pa, sin pa)
    const float pb =  0.5f * (phi - om);     // em = (cos pb, sin pb)
    const float ca = cosf(pa), sa = sinf(pa);
    const float cb = cosf(pb), sb = sinf(pb);
    s_rot[tid][0] =  ca * ch;  s_rot[tid][1] =  sa * ch;   // u00
    s_rot[tid][2] = -cb * sh;  s_rot[tid][3] = -sb * sh;   // u01
    s_rot[tid][4] =  cb * sh;  s_rot[tid][5] = -sb * sh;   // u10
    s_rot[tid][6] =  ca * ch;  s_rot[tid][7] = -sa * ch;   // u11
  }
  __syncthreads();

  const int tok0 = blockIdx.x * 64 + wv * 16;
  const float* xrow = x + (size_t)(tok0 + col) * 128;
  __builtin_prefetch(xrow, 0, 0);            // global_prefetch_b8

  // ---- encoder GEMM1: h = x(16x128) @ w1(128x64), 128 wmma ------------------
  v8f acc[4];
#pragma unroll
  for (int n = 0; n < 4; ++n) acc[n] = (v8f){0.f,0.f,0.f,0.f,0.f,0.f,0.f,0.f};
#pragma unroll
  for (int k = 0; k < 32; ++k) {
    const v2f a = *(const v2f*)(xrow + 4 * k + r0);
#pragma unroll
    for (int n = 0; n < 4; ++n) {
      const v2f b = *(const v2f*)(&s_w1t[16 * n + col][4 * k + r0]);
      acc[n] = wmma4(a, b, acc[n]);
    }
  }
  // bias + exact GELU, stage h tile for re-layout as A operand
#pragma unroll
  for (int n = 0; n < 4; ++n) {
    const float bv = s_b1[16 * n + col];
#pragma unroll
    for (int r = 0; r < 8; ++r) {
      s_h[wv][r + 8 * hi][16 * n + col] = gelu_exact(acc[n][r] + bv);
    }
  }

  // ---- encoder GEMM2: ang = h(16x64) @ w2(64x8), 16 wmma (N padded) ---------
  v8f eacc = (v8f){0.f,0.f,0.f,0.f,0.f,0.f,0.f,0.f};
#pragma unroll
  for (int k = 0; k < 16; ++k) {
    const v2f a = *(const v2f*)(&s_h[wv][col][4 * k + r0]);
    float b0 = 0.0f, b1 = 0.0f;
    if (col < 8) { b0 = s_w2t[col][4 * k + r0]; b1 = s_w2t[col][4 * k + r0 + 1]; }
    const v2f b = {b0, b1};
    eacc = wmma4(a, b, eacc);
  }
  if (col < 8) {
    const float bv = s_b2[col];
#pragma unroll
    for (int r = 0; r < 8; ++r) s_ang[wv][r + 8 * hi][col] = eacc[r] + bv;
  }

  // ---- 4-qubit statevector simulation, one token per lane (lanes 0-15) ------
  if (lane < 16) {
    float zr[16], zi[16];
#pragma unroll
    for (int s = 0; s < 16; ++s) { zr[s] = 0.0f; zi[s] = 0.0f; }
    zr[0] = 1.0f;

    // encoding: RX(ang[2i]) RY(ang[2i+1]) on qubit i (qubit i -> bit mask 8>>i)
#pragma unroll
    for (int i = 0; i < 4; ++i) {
      const int m = 8 >> i;
      const float tx = s_ang[wv][lane][2 * i];
      const float ty = s_ang[wv][lane][2 * i + 1];
      float c = cosf(0.5f * tx), sn = sinf(0.5f * tx);
      apply1q(zr, zi, m, c, 0.f, 0.f, -sn, 0.f, -sn, c, 0.f);     // RX
      c = cosf(0.5f * ty); sn = sinf(0.5f * ty);
      apply1q(zr, zi, m, c, 0.f, -sn, 0.f, sn, 0.f, c, 0.f);      // RY
    }
    // variational layers: Rot per qubit + CNOT ring
#pragma unroll
    for (int l = 0; l < 3; ++l) {
#pragma unroll
      for (int i = 0; i < 4; ++i) {
        const float* u = s_rot[l * 4 + i];
        apply1q(zr, zi, 8 >> i, u[0], u[1], u[2], u[3], u[4], u[5], u[6], u[7]);
      }
#pragma unroll
      for (int i = 0; i < 4; ++i) {
        const int mc = 8 >> i;
        const int mt = 8 >> ((i + 1) & 3);
#pragma unroll
        for (int s = 0; s < 16; ++s) {
          if ((s & mc) && !(s & mt)) {
            const int t = s | mt;
            float tr = zr[s]; zr[s] = zr[t]; zr[t] = tr;
            float ti = zi[s]; zi[s] = zi[t]; zi[t] = ti;
          }
        }
      }
    }
    float p[16];
#pragma unroll
    for (int s = 0; s < 16; ++s) p[s] = zr[s] * zr[s] + zi[s] * zi[s];
#pragma unroll
    for (int i = 0; i < 4; ++i) {
      const int m = 8 >> i;
      float z = 0.0f;
#pragma unroll
      for (int s = 0; s < 16; ++s) z += (s & m) ? -p[s] : p[s];
      s_qf[wv][lane][i] = z;
      const float qp = (z + 1.0f) * 0.5f;
      const float et = -(qp * logf(qp + 1e-10f) +
                         (1.0f - qp) * logf(1.0f - qp + 1e-10f));
      atomicAdd(&s_ent[i], et);                    // ds_add_f32
    }
  }

  // ---- decoder GEMM1: g = qf(16x4) @ dw1(4x16), 1 wmma ----------------------
  {
    const v2f a = *(const v2f*)(&s_qf[wv][col][r0]);
    const v2f b = *(const v2f*)(&s_dw1t[col][r0]);
    v8f g = (v8f){0.f,0.f,0.f,0.f,0.f,0.f,0.f,0.f};
    g = wmma4(a, b, g);
    const float bv = s_db1[col];
#pragma unroll
    for (int r = 0; r < 8; ++r)
      s_g[wv][r + 8 * hi][col] = gelu_exact(g[r] + bv);
  }

  // ---- decoder GEMM2: latent = g(16x16) @ dw2(16x64), 16 wmma ---------------
  v8f oacc[4];
#pragma unroll
  for (int n = 0; n < 4; ++n) oacc[n] = (v8f){0.f,0.f,0.f,0.f,0.f,0.f,0.f,0.f};
#pragma unroll
  for (int k = 0; k < 4; ++k) {
    const v2f a = *(const v2f*)(&s_g[wv][col][4 * k + r0]);
#pragma unroll
    for (int n = 0; n < 4; ++n) {
      const v2f b = *(const v2f*)(&s_dw2t[16 * n + col][4 * k + r0]);
      oacc[n] = wmma4(a, b, oacc[n]);
    }
  }
  float* obase = out + (size_t)tok0 * 64;
#pragma unroll
  for (int n = 0; n < 4; ++n) {
    const float bv = s_db2[16 * n + col];
#pragma unroll
    for (int r = 0; r < 8; ++r) {
      __builtin_nontemporal_store(
          oacc[n][r] + bv,
          obase + (size_t)(r + 8 * hi) * 64 + 16 * n + col);
    }
  }

  // ---- entropy: one global atomic per accumulator per block -----------------
  __syncthreads();
  if (tid < 4) atomicAdd(ent + tid, s_ent[tid] * (1.0f / (float)TOK_TOTAL));
}

extern "C" void kernel_launch(void* const* d_in, const int* in_sizes, int n_in,
                              void* d_out, int out_size, void* d_ws, size_t ws_size,
                              hipStream_t stream) {
  (void)in_sizes; (void)n_in; (void)out_size; (void)d_ws; (void)ws_size;
  const float* x        = (const float*)d_in[0];
  const float* enc_w1   = (const float*)d_in[1];
  const float* enc_b1   = (const float*)d_in[2];
  const float* enc_w2   = (const float*)d_in[3];
  const float* enc_b2   = (const float*)d_in[4];
  const float* q_params = (const float*)d_in[5];
  const float* dec_w1   = (const float*)d_in[6];
  const float* dec_b1   = (const float*)d_in[7];
  const float* dec_w2   = (const float*)d_in[8];
  const float* dec_b2   = (const float*)d_in[9];

  float* out = (float*)d_out;
  float* ent = out + LATENT_ELEMS;   // 4 entropy scalars after latent

  qvl_zero_ent<<<1, 32, 0, stream>>>(ent);
  qvl_fused<<<TOK_TOTAL / 64, 128, 0, stream>>>(
      x, enc_w1, enc_b1, enc_w2, enc_b2, q_params,
      dec_w1, dec_b1, dec_w2, dec_b2, out, ent);
}